// Dynamic_conv1d_44633300140754
// MI455X (gfx1250) — compile-verified
//
#include <hip/hip_runtime.h>

typedef __attribute__((ext_vector_type(2))) float v2f;
typedef __attribute__((ext_vector_type(8))) float v8f;

#define B_    4
#define C_    256
#define HW_   (192 * 192)       // 36864, divisible by 16
#define K_    4
#define HID_  16
#define NTILE (HW_ / 16)        // 2304 pixel tiles per batch
#define INV_T (1.0f / 34.0f)

// ---------------- Kernel 1: wsum[k][i] = sum_o weight[k][o][i] ----------------
__global__ void __launch_bounds__(256) wsum_kernel(const float* __restrict__ weight,
                                                   float* __restrict__ wsum) {
    int tid = blockIdx.x * 256 + threadIdx.x;   // 0..1023
    int k = tid >> 8;
    int i = tid & 255;
    const float* wp = weight + (size_t)k * C_ * C_ + i;
    float s = 0.0f;
#pragma unroll 8
    for (int o = 0; o < C_; ++o) s += wp[(size_t)o * C_];
    wsum[tid] = s;
}

// ---------------- Kernel 2: streaming pass, WMMA 16x16x4 for scale ----------------
// wave  -> one 16-pixel tile; loops over 16 channel tiles of 16
// A (16x4)  = w_sum^T[c0+0..15, k0..3]   lanes 0-15: K=0,1 ; lanes 16-31: K=2,3
// B (4x16)  = attn^T[k0..3, p0+0..15]    lanes 0-15: K=0,1 ; lanes 16-31: K=2,3
// D (16x16) = scale[c0+r+8*hi, p0+(lane&15)] in VGPR r
__global__ void __launch_bounds__(256) dynconv_kernel(
    const float* __restrict__ x, const float* __restrict__ qmap,
    const float* __restrict__ fc1w, const float* __restrict__ fc2w,
    const float* __restrict__ fc2b, const float* __restrict__ wsum,
    float* __restrict__ out)
{
    const int lane = threadIdx.x & 31;
    const int wave = threadIdx.x >> 5;
    const int tile = blockIdx.x * 8 + wave;          // 0 .. 9215
    const int b    = tile / NTILE;
    const int p0   = (tile % NTILE) * 16;
    const int lp   = lane & 15;                      // pixel within tile
    const int hi   = lane >> 4;                      // half-wave select (K pair)

    // ---- per-pixel attention (each lane handles pixel p0+lp; upper half duplicates) ----
    const float q = qmap[(size_t)b * HW_ + p0 + lp];
    float z0 = fc2b[0], z1 = fc2b[1], z2 = fc2b[2], z3 = fc2b[3];
#pragma unroll
    for (int o = 0; o < HID_; ++o) {
        float h = q * fc1w[o];
        h = h > 0.0f ? h : 0.0f;                     // relu
        z0 += fc2w[0 * HID_ + o] * h;
        z1 += fc2w[1 * HID_ + o] * h;
        z2 += fc2w[2 * HID_ + o] * h;
        z3 += fc2w[3 * HID_ + o] * h;
    }
    z0 *= INV_T; z1 *= INV_T; z2 *= INV_T; z3 *= INV_T;
    float m = fmaxf(fmaxf(z0, z1), fmaxf(z2, z3));
    float e0 = __expf(z0 - m), e1 = __expf(z1 - m);
    float e2 = __expf(z2 - m), e3 = __expf(z3 - m);
    float rs = __builtin_amdgcn_rcpf(e0 + e1 + e2 + e3);   // fast v_rcp_f32
    float a0 = e0 * rs, a1 = e1 * rs, a2 = e2 * rs, a3 = e3 * rs;

    // B operand: this lane's pixel, K = 2*hi and 2*hi+1
    v2f Bm;
    Bm.x = hi ? a2 : a0;
    Bm.y = hi ? a3 : a1;

    const size_t xbase = (size_t)b * C_ * HW_ + p0 + lp;
    const float* wsA = wsum + 2 * hi * C_ + lp;      // rows K=2*hi, K=2*hi+1

#pragma unroll 4
    for (int ct = 0; ct < 16; ++ct) {
        const int c0 = ct * 16;
        // A operand: w_sum^T tile, channel row = c0 + lp
        v2f Am;
        Am.x = wsA[c0];                              // K = 2*hi
        Am.y = wsA[C_ + c0];                         // K = 2*hi + 1
        v8f acc = {};
        acc = __builtin_amdgcn_wmma_f32_16x16x4_f32(
            /*neg_a=*/false, Am, /*neg_b=*/false, Bm,
            /*c_mod=*/(short)0, acc, /*reuse_a=*/false, /*reuse_b=*/false);

        // D[r] = scale for channel c0 + r + 8*hi at pixel p0+lp : coalesced rows of 16
        // out = x + x*scale -> single v_fma per element
#pragma unroll
        for (int r = 0; r < 8; ++r) {
            const size_t off = xbase + (size_t)(c0 + r + 8 * hi) * HW_;
            const float xv = __builtin_nontemporal_load(x + off);
            __builtin_nontemporal_store(xv + xv * acc[r], out + off);
        }
    }
}

extern "C" void kernel_launch(void* const* d_in, const int* in_sizes, int n_in,
                              void* d_out, int out_size, void* d_ws, size_t ws_size,
                              hipStream_t stream) {
    const float* x      = (const float*)d_in[0];
    const float* qmap   = (const float*)d_in[1];
    const float* fc1w   = (const float*)d_in[2];
    const float* fc2w   = (const float*)d_in[3];
    const float* fc2b   = (const float*)d_in[4];
    const float* weight = (const float*)d_in[5];
    float* out  = (float*)d_out;
    float* wsum = (float*)d_ws;                      // 4*256 floats = 4 KB

    wsum_kernel<<<4, 256, 0, stream>>>(weight, wsum);
    dynconv_kernel<<<(B_ * NTILE) / 8, 256, 0, stream>>>(x, qmap, fc1w, fc2w,
                                                         fc2b, wsum, out);
}